// Attention_67714454389365
// MI455X (gfx1250) — compile-verified
//
#include <hip/hip_runtime.h>
#include <math.h>

// ---------------------------------------------------------------------------
// Problem constants (from the reference)
// ---------------------------------------------------------------------------
#define D_DIM   512
#define B_IMGS  4
#define N_NETS  4
#define C_CH    64
#define HH      256
#define WW      256
#define RES_ELEMS (B_IMGS * C_CH * HH * WW)         // 16,777,216 floats
#define PLANE     (C_CH * HH * WW)                  // 4,194,304 floats per (b,n)
#define PLANE4    (PLANE / 4)                       // 1,048,576 float4 per plane
#define INV_T   0.04419417382415922f                // 1/sqrt(512)

typedef __attribute__((ext_vector_type(2))) float v2f;
typedef __attribute__((ext_vector_type(4))) float f4;
typedef __attribute__((ext_vector_type(8))) float v8f;

// ---------------------------------------------------------------------------
// Kernel 1: attn = (q/T) @ k^T via V_WMMA_F32_16X16X4_F32, then softmax.
// One wave32. A-matrix rows m = q[m&3] (replication of unused rows is
// harmless: only D rows 0..3 are read out). B-matrix columns n16 = b*4+n are
// the 16 flattened k rows. Accumulate over D=512 in 128 chunks of K=4.
//
// A 16x4 f32 layout: lane L<16 -> M=L, K={0,1} in v[0],v[1];
//                    lane L>=16 -> M=L-16, K={2,3}.
// B 4x16  f32 layout: lane L<16 -> N=L, K={0,1}; lane L>=16 -> N=L-16, K={2,3}.
// D 16x16 f32 layout: VGPR r -> M=r (lanes 0-15, N=lane) / M=r+8 (lanes 16-31).
// Wanted entry D[b][b*4+n] lives in element b of lane (b*4+n).
// ---------------------------------------------------------------------------
__launch_bounds__(32)
__global__ void attn_softmax_kernel(const float* __restrict__ q,   // [4,512]
                                    const float* __restrict__ k,   // [16,512]
                                    float* __restrict__ attn_out,  // [16]
                                    float* __restrict__ w_out) {   // [16]
    __shared__ float s_attn[16];

    const int lane = threadIdx.x;       // 0..31, full wave active (EXEC all 1s)
    const int mrow = lane & 15;
    const int hiK  = lane >> 4;         // 0 -> K{0,1}, 1 -> K{2,3}

    const float* qrow = q + (size_t)(mrow & 3) * D_DIM;  // replicate q rows
    const float* krow = k + (size_t)mrow * D_DIM;        // n16 = mrow

    v8f c = {};
#pragma unroll 8
    for (int kc = 0; kc < D_DIM; kc += 4) {
        const int off = kc + hiK * 2;                    // 8-byte aligned
        const v2f a = *reinterpret_cast<const v2f*>(qrow + off);
        const v2f b = *reinterpret_cast<const v2f*>(krow + off);
        // D = A*B + C ; args: (neg_a, A, neg_b, B, c_mod, C, reuse_a, reuse_b)
        c = __builtin_amdgcn_wmma_f32_16x16x4_f32(false, a, false, b,
                                                  (short)0, c, false, false);
    }

    if (lane < 16) {
        const int bq = lane >> 2;       // image index
        float dot = c[0];
        if (bq == 1) dot = c[1];
        if (bq == 2) dot = c[2];
        if (bq == 3) dot = c[3];
        s_attn[lane] = dot * INV_T;     // pre-softmax attn (second output)
    }
    __syncthreads();

    if (lane < 16) {
        const int bq = lane >> 2;
        const float a0 = s_attn[bq * 4 + 0];
        const float a1 = s_attn[bq * 4 + 1];
        const float a2 = s_attn[bq * 4 + 2];
        const float a3 = s_attn[bq * 4 + 3];
        const float mx = fmaxf(fmaxf(a0, a1), fmaxf(a2, a3));
        const float sum = __expf(a0 - mx) + __expf(a1 - mx) +
                          __expf(a2 - mx) + __expf(a3 - mx);
        const float mine = s_attn[lane];
        w_out[lane]    = __expf(mine - mx) / sum;  // softmax weight
        attn_out[lane] = mine;                     // raw scaled logits
    }
}

// ---------------------------------------------------------------------------
// Kernel 2: res[b,:,:,:] = sum_n w[b,n] * v[b,n,:,:,:].
// Pure HBM-streaming: 268 MB read + 67 MB write -> ~14.4 us floor at 23.3 TB/s.
// float4 (b128) loads/stores, non-temporal (no reuse; 268 MB > 192 MB L2).
// grid = (PLANE4/256, B); one float4 per thread -> 4M threads in flight.
// ---------------------------------------------------------------------------
__launch_bounds__(256)
__global__ void weighted_sum_kernel(const float* __restrict__ v,
                                    const float* __restrict__ w,
                                    float* __restrict__ out) {
    const int b = blockIdx.y;
    const size_t i = (size_t)blockIdx.x * blockDim.x + threadIdx.x;  // < PLANE4

    const float w0 = w[b * 4 + 0];
    const float w1 = w[b * 4 + 1];
    const float w2 = w[b * 4 + 2];
    const float w3 = w[b * 4 + 3];

    const f4* vb = reinterpret_cast<const f4*>(v) + (size_t)b * N_NETS * PLANE4;
    const f4 x0 = __builtin_nontemporal_load(vb + 0 * (size_t)PLANE4 + i);
    const f4 x1 = __builtin_nontemporal_load(vb + 1 * (size_t)PLANE4 + i);
    const f4 x2 = __builtin_nontemporal_load(vb + 2 * (size_t)PLANE4 + i);
    const f4 x3 = __builtin_nontemporal_load(vb + 3 * (size_t)PLANE4 + i);

    f4 r;
    r.x = fmaf(w0, x0.x, fmaf(w1, x1.x, fmaf(w2, x2.x, w3 * x3.x)));
    r.y = fmaf(w0, x0.y, fmaf(w1, x1.y, fmaf(w2, x2.y, w3 * x3.y)));
    r.z = fmaf(w0, x0.z, fmaf(w1, x1.z, fmaf(w2, x2.z, w3 * x3.z)));
    r.w = fmaf(w0, x0.w, fmaf(w1, x1.w, fmaf(w2, x2.w, w3 * x3.w)));

    __builtin_nontemporal_store(r, reinterpret_cast<f4*>(out) + (size_t)b * PLANE4 + i);
}

// ---------------------------------------------------------------------------
// Entry point. Inputs: d_in[0]=q [4,1,1,512] f32, d_in[1]=k [4,1,4,512] f32,
// d_in[2]=v [4,4,64,256,256] f32. Output: res (16,777,216 f32) ++ attn (16 f32).
// Workspace: first 16 floats hold the softmax weights (written every call).
// ---------------------------------------------------------------------------
extern "C" void kernel_launch(void* const* d_in, const int* in_sizes, int n_in,
                              void* d_out, int out_size, void* d_ws, size_t ws_size,
                              hipStream_t stream) {
    const float* q = (const float*)d_in[0];
    const float* k = (const float*)d_in[1];
    const float* v = (const float*)d_in[2];

    float* res  = (float*)d_out;                 // first output
    float* attn = (float*)d_out + RES_ELEMS;     // second output (16 floats)
    float* w    = (float*)d_ws;                  // scratch: softmax weights

    attn_softmax_kernel<<<dim3(1), dim3(32), 0, stream>>>(q, k, attn, w);

    weighted_sum_kernel<<<dim3(PLANE4 / 256, B_IMGS), dim3(256), 0, stream>>>(
        v, w, res);
}